// IDM_43748536877069
// MI455X (gfx1250) — compile-verified
//
#include <hip/hip_runtime.h>
#include <math.h>

// IDM pairwise leader-search + action. The O(NV^2) part is rank-3 affine
// projections computed with V_WMMA_F32_16X16X4_F32 (fp32-exact):
//
//   ndist[i,j] = (x_i-x_j)cos(psi_j) + (y_i-y_j)sin(psi_j)
//              =  x_i*a_j + y_i*b_j + c_j      (a=cos, b=sin, c=-(x_j a + y_j b))
//   tdist[i,j] = -(x_i-x_j)sin(psi_j) + (y_i-y_j)cos(psi_j)
//   cone       =  |tdist| < ndist*tan(20deg)  &&  i != j
//     (|t| < n*tan20 already implies n > 0: if n <= 0 then n*tan20 <= 0 <= |t|)
//
// Reduction is fully branchless (v_cmp + v_cndmask), no EXEC manipulation.

typedef float v2f __attribute__((ext_vector_type(2)));
typedef float v8f __attribute__((ext_vector_type(8)));

#define NVEH 2048
#define THREADS 256
#define NWAVES 8
#define ROWS_PER_WAVE (NVEH / NWAVES)   // 256
#define NCHUNKS (ROWS_PER_WAVE / 16)    // 16

__global__ __launch_bounds__(THREADS)
void idm_wmma_kernel(const float* __restrict__ state,    // [B, NVEH*5]
                     const float* __restrict__ lengths,  // [NVEH]
                     const float* __restrict__ v0p,
                     const float* __restrict__ s0p,
                     const float* __restrict__ dthp,
                     const float* __restrict__ amaxp,
                     const float* __restrict__ bp,
                     float* __restrict__ out)            // [B, NVEH]
{
    __shared__ float sx[NVEH];
    __shared__ float sy[NVEH];
    __shared__ float s_pmin[NWAVES][16];
    __shared__ int   s_pidx[NWAVES][16];

    const int tid    = threadIdx.x;
    const int lane   = tid & 31;           // wave32
    const int wave   = tid >> 5;
    const int hi     = (lane >> 4) & 1;    // half-wave (rows 0-7 vs 8-15 of C/D)
    const int col16  = lane & 15;          // column within the 16x16 tile
    const bool lanelo = (lane < 16);
    const int b      = blockIdx.y;
    const int colBase = blockIdx.x * 16;
    const int j = colBase + col16;         // this lane's ego column

    const float* __restrict__ stb = state + (size_t)b * NVEH * 5;

    // ---- stage candidate (x,y) into LDS once per block ----
    for (int k = tid; k < NVEH; k += THREADS) {
        sx[k] = stb[k * 5 + 0];
        sy[k] = stb[k * 5 + 1];
    }
    __syncthreads();

    // ---- per-ego-column setup: ALL lanes load column j (hi half duplicates
    //      the lo half's addresses -> cache broadcast, no divergence) ----
    const float* pj = stb + (size_t)j * 5;
    const float xj = pj[0], yj = pj[1], vj = pj[2], psij = pj[3];
    const float cp = cosf(psij);
    const float sp = sinf(psij);

    // bias terms valid in every lane (lane L and L+16 hold the same column)
    const float cn = -(xj * cp + yj * sp);   // ndist bias
    const float ct =  (xj * sp - yj * cp);   // tdist bias
    v8f Cn, Ct;
#pragma unroll
    for (int q = 0; q < 8; ++q) { Cn[q] = cn; Ct[q] = ct; }

    // B matrices: K0,K1 rows live in lanes 0..15; K2,K3 rows must be zero
    v2f Bn, Bt;
    Bn.x = lanelo ?  cp : 0.0f;  Bn.y = lanelo ? sp : 0.0f;
    Bt.x = lanelo ? -sp : 0.0f;  Bt.y = lanelo ? cp : 0.0f;

    const float TAN_HALF = 0.36397023426620234f;  // tan(20 deg)
    float best = __builtin_inff();
    int   bidx = 0;

    const int rowBase0 = wave * ROWS_PER_WAVE;
    for (int c = 0; c < NCHUNKS; ++c) {
        const int rowBase = rowBase0 + c * 16;

        // A matrix: lanes 0..15 hold {x_i, y_i} (K0,K1); hi half -> 0 (K2,K3)
        // all lanes read the same LDS row -> no EXEC-guarded ds_load
        const int arow = rowBase + col16;
        const float ax = sx[arow];
        const float ay = sy[arow];
        v2f A;
        A.x = lanelo ? ax : 0.0f;
        A.y = lanelo ? ay : 0.0f;

        v8f dn = __builtin_amdgcn_wmma_f32_16x16x4_f32(
            false, A, false, Bn, (short)0, Cn, false, false);
        v8f dt = __builtin_amdgcn_wmma_f32_16x16x4_f32(
            false, A, false, Bt, (short)0, Ct, false, false);

        const int ibase = rowBase + (hi << 3);  // this lane's first row index
        const int selfq = j - ibase;            // q that would be self (may be OOR)

#pragma unroll
        for (int q = 0; q < 8; ++q) {
            const float n = dn[q];
            const float t = dt[q];
            // |t| < n*tan20 subsumes n > 0; branchless combined lane mask
            const bool better = (fabsf(t) < n * TAN_HALF)
                              & (q != selfq) & (n < best);
            best = better ? n : best;
            bidx = better ? (ibase + q) : bidx;
        }
    }

    // ---- combine the two half-wave row groups (hi half = larger indices) ----
    {
        const float ov = __shfl_xor(best, 16, 32);
        const int   oi = __shfl_xor(bidx, 16, 32);
        const bool take = ov < best;            // strict: keeps lower index on tie
        best = take ? ov : best;
        bidx = take ? oi : bidx;
    }
    if (lanelo) { s_pmin[wave][col16] = best; s_pidx[wave][col16] = bidx; }
    __syncthreads();

    // ---- final cross-wave reduce + IDM action (threads 0..15 == wave0 lanes) ----
    if (tid < 16) {
        float bv = s_pmin[0][tid];
        int   bi = s_pidx[0][tid];
#pragma unroll
        for (int w = 1; w < NWAVES; ++w) {      // ascending wave = ascending rows
            const float pv = s_pmin[w][tid];
            const bool take = pv < bv;
            bv = take ? pv : bv;
            bi = take ? s_pidx[w][tid] : bi;
        }

        const float v0   = v0p[0];
        const float s0   = s0p[0];
        const float dth  = dthp[0];
        const float amax = amaxp[0];
        const float bb   = bp[0];

        const float sal    = bv - lengths[j];
        const bool  finite = isfinite(sal);
        const float salSafe = finite ? sal : 1.0f;

        // leading vehicle velocity, projected onto ego heading (exact ndv)
        const float* pl = stb + (size_t)bi * 5;
        const float vl = pl[2], psil = pl[3];
        const float dvx = vl * cosf(psil) - vj * cp;
        const float dvy = vl * sinf(psil) - vj * sp;
        const float ndv = dvx * cp + dvy * sp;

        const float sstar = s0 + vj * dth + vj * ndv / (2.0f * sqrtf(amax * bb));
        const float tr = vj / v0;
        const float t2 = tr * tr;
        const float actFree = amax * (1.0f - t2 * t2);
        const float ratio = sstar / salSafe;
        const float act = finite ? (actFree - amax * ratio * ratio) : actFree;

        out[(size_t)b * NVEH + j] = act;
    }
}

extern "C" void kernel_launch(void* const* d_in, const int* in_sizes, int n_in,
                              void* d_out, int out_size, void* d_ws, size_t ws_size,
                              hipStream_t stream) {
    const float* state   = (const float*)d_in[0];  // [B, NVEH*5]
    const float* lengths = (const float*)d_in[1];  // [NVEH]
    const float* v0      = (const float*)d_in[2];
    const float* s0      = (const float*)d_in[3];
    const float* dth     = (const float*)d_in[4];
    const float* amax    = (const float*)d_in[5];
    const float* bcoef   = (const float*)d_in[6];
    float* out = (float*)d_out;

    const int B = in_sizes[0] / (NVEH * 5);        // 8
    dim3 grid(NVEH / 16, B);                       // 128 x 8 = 1024 blocks
    idm_wmma_kernel<<<grid, THREADS, 0, stream>>>(state, lengths, v0, s0, dth,
                                                  amax, bcoef, out);
}